// MANN_LSTMCell_2104533975859
// MI455X (gfx1250) — compile-verified
//
#include <hip/hip_runtime.h>
#include <hip/hip_bf16.h>

typedef __attribute__((ext_vector_type(2))) float v2f;
typedef __attribute__((ext_vector_type(8))) float v8f;

#define B_   32
#define U_   1024
#define M_   65536
#define FU_  4096      // 4*U
#define RKLD 5120      // 5*U (recurrent_kernel leading dim)

__device__ __forceinline__ v8f wmma_f32(v2f a, v2f b, v8f c) {
    // D = A(16x4 f32) * B(4x16 f32) + C(16x16 f32)
    return __builtin_amdgcn_wmma_f32_16x16x4_f32(false, a, false, b, (short)0, c,
                                                 false, false);
}

// Async DMA: 16 bytes global -> LDS, tracked by ASYNCcnt (CDNA5 §15.18.3).
__device__ __forceinline__ void async_ld_b128(const float* g, unsigned int lds_off) {
    asm volatile("global_load_async_to_lds_b128 %0, %1, off"
                 :: "v"(lds_off), "v"(g) : "memory");
}

__device__ __forceinline__ float hsig(float x) {
    return fminf(fmaxf(0.2f * x + 0.5f, 0.0f), 1.0f);
}

// ---------------------------------------------------------------- init
__global__ void k_init(float* __restrict__ r_out, float* __restrict__ rowsum,
                       int* __restrict__ sel) {
    int t = blockIdx.x * blockDim.x + threadIdx.x;
    if (t < B_ * U_) r_out[t] = 0.0f;
    if (t < B_) rowsum[t] = 0.0f;
    if (t == 0) *sel = 0x7fffffff;
}

// ------------------------------------------------- zcat = x@W + h@RK4 + bias
// 512 wave-tiles: (2 M-tiles) x (256 N-tiles), K = 1024
__global__ void k_gates(const float* __restrict__ x, const float* __restrict__ h_tm1,
                        const float* __restrict__ W, const float* __restrict__ RK,
                        const float* __restrict__ bias, float* __restrict__ zcat) {
    int g    = (blockIdx.x * blockDim.x + threadIdx.x) >> 5;
    int lane = threadIdx.x & 31;
    int ln = lane & 15, hh = lane >> 4;
    int nt = g & 255, mt = g >> 8;
    int m0 = mt * 16, n0 = nt * 16;
    v8f acc = {};
    for (int k = 0; k < U_; k += 4) {
        int ka = k + 2 * hh;
        v2f a1 = *(const v2f*)(x + (m0 + ln) * U_ + ka);
        v2f b1; b1.x = W[ka * FU_ + n0 + ln]; b1.y = W[(ka + 1) * FU_ + n0 + ln];
        acc = wmma_f32(a1, b1, acc);
        v2f a2 = *(const v2f*)(h_tm1 + (m0 + ln) * U_ + ka);
        v2f b2; b2.x = RK[ka * RKLD + n0 + ln]; b2.y = RK[(ka + 1) * RKLD + n0 + ln];
        acc = wmma_f32(a2, b2, acc);
    }
    float bv = bias[n0 + ln];
    for (int j = 0; j < 8; ++j)
        zcat[(m0 + j + 8 * hh) * FU_ + n0 + ln] = acc[j] + bv;
}

// ------------------------------------------------- ri = r_tm1 @ RK[:, 4U:]
// 128 wave-tiles: (2 M-tiles) x (64 N-tiles), K = 1024
__global__ void k_ri(const float* __restrict__ r_tm1, const float* __restrict__ RK,
                     float* __restrict__ ri) {
    int g    = (blockIdx.x * blockDim.x + threadIdx.x) >> 5;
    int lane = threadIdx.x & 31;
    int ln = lane & 15, hh = lane >> 4;
    int nt = g & 63, mt = g >> 6;
    int m0 = mt * 16, n0 = nt * 16;
    v8f acc = {};
    for (int k = 0; k < U_; k += 4) {
        int ka = k + 2 * hh;
        v2f a = *(const v2f*)(r_tm1 + (m0 + ln) * U_ + ka);
        v2f b; b.x = RK[ka * RKLD + FU_ + n0 + ln];
               b.y = RK[(ka + 1) * RKLD + FU_ + n0 + ln];
        acc = wmma_f32(a, b, acc);
    }
    for (int j = 0; j < 8; ++j)
        ri[(m0 + j + 8 * hh) * U_ + n0 + ln] = acc[j];
}

// ---------------------------------------------------------------- LSTM gates
__global__ void k_lstm(const float* __restrict__ zcat, const float* __restrict__ ri,
                       const float* __restrict__ c_tm1, float* __restrict__ c_out,
                       float* __restrict__ h_out) {
    int t = blockIdx.x * blockDim.x + threadIdx.x;   // B*U
    int b = t >> 10, u = t & (U_ - 1);
    float zi = zcat[b * FU_ + u];
    float zf = zcat[b * FU_ + U_ + u];
    float zc = zcat[b * FU_ + 2 * U_ + u];
    float zo = zcat[b * FU_ + 3 * U_ + u];
    float i = hsig(zi + ri[t]);
    float f = hsig(zf);
    float c = f * c_tm1[t] + i * tanhf(zc);
    float o = hsig(zo);
    c_out[t] = c;
    h_out[t] = o * tanhf(c);
}

// ---------------------------------------------------------------- |h| row norms
__global__ void k_hnorm(const float* __restrict__ h, float* __restrict__ hinv) {
    __shared__ float sd[256];
    int b = blockIdx.x;
    float s = 0.0f;
    for (int u = threadIdx.x; u < U_; u += 256) { float v = h[b * U_ + u]; s += v * v; }
    sd[threadIdx.x] = s;
    __syncthreads();
    for (int o = 128; o > 0; o >>= 1) {
        if (threadIdx.x < o) sd[threadIdx.x] += sd[threadIdx.x + o];
        __syncthreads();
    }
    if (threadIdx.x == 0) hinv[b] = rsqrtf(fmaxf(sd[0], 1e-12f));
}

// ------------------------ sim chunk: exp(cos-sim) + fused bank row norms
// grid = 4096 chunks of 16 bank rows; block = 64 (2 waves, one per batch half).
// Bank rows are DMA'd into LDS with global_load_async_to_lds_b128 (double
// buffered, ASYNCcnt), then read as WMMA A-fragments via ds_load (stride 68
// floats: 16B-aligned rows, conflict-free banks 4r+c).
#define SIM_KBLK 64
#define SIM_STR  68
__global__ void k_sim(const float* __restrict__ mb, const float* __restrict__ h,
                      const float* __restrict__ hinv, float* __restrict__ wr_exp,
                      float* __restrict__ rowsum) {
    __shared__ float sA[2][16 * SIM_STR];   // 2 x 4352 B
    int chunk = blockIdx.x;
    int tid   = threadIdx.x;               // 0..63
    int wv    = tid >> 5;                  // batch half (0/1)
    int lane  = tid & 31;
    int ln = lane & 15, hh = lane >> 4;
    int n0 = wv * 16;

    const float* gbase = mb + (size_t)chunk * 16 * U_;

    // stage 0
    for (int i = 0; i < 4; ++i) {
        int u = i * 64 + tid;              // 256 x 16B units per stage
        int row = u >> 4, c4 = (u & 15) * 4;
        async_ld_b128(gbase + (size_t)row * U_ + c4,
                      (unsigned int)(unsigned long long)&sA[0][row * SIM_STR + c4]);
    }

    v8f acc = {};
    float ss = 0.0f;
    for (int kt = 0; kt < U_ / SIM_KBLK; ++kt) {
        int buf = kt & 1;
        if (kt + 1 < U_ / SIM_KBLK) {
            int k0n = (kt + 1) * SIM_KBLK;
            for (int i = 0; i < 4; ++i) {
                int u = i * 64 + tid;
                int row = u >> 4, c4 = (u & 15) * 4;
                async_ld_b128(gbase + (size_t)row * U_ + k0n + c4,
                              (unsigned int)(unsigned long long)
                                  &sA[buf ^ 1][row * SIM_STR + c4]);
            }
            asm volatile("s_wait_asynccnt 0x4" ::: "memory");  // stage kt done
        } else {
            asm volatile("s_wait_asynccnt 0x0" ::: "memory");
        }
        __syncthreads();
        const float* As = &sA[buf][0];
        const float* Hs = h + (n0 + ln) * U_ + kt * SIM_KBLK;
        for (int k = 0; k < SIM_KBLK; k += 4) {
            int ka = k + 2 * hh;
            v2f a = *(const v2f*)(As + ln * SIM_STR + ka);
            ss += a.x * a.x + a.y * a.y;    // fused bank row sum-of-squares
            v2f b = *(const v2f*)(Hs + ka);
            acc = wmma_f32(a, b, acc);
        }
        __syncthreads();
    }

    ss += __shfl_xor(ss, 16, 32);                     // combine lane halves
    float inv = rsqrtf(fmaxf(ss, 1e-12f));            // per bank-row inv norm
    float hi  = hinv[n0 + ln];                        // batch col inv norm
    float psum = 0.0f;
    for (int j = 0; j < 8; ++j) {
        int mr = j + 8 * hh;
        float im = __shfl(inv, mr, 32);
        float e  = __expf(acc[j] * im * hi);          // cos-sim <= 1: exp safe
        wr_exp[(size_t)(n0 + ln) * M_ + chunk * 16 + mr] = e;
        psum += e;
    }
    atomicAdd(&rowsum[n0 + ln], psum);
}

// ------------------------ normalize wr, ww on the fly, wu_new
__global__ void k_normwu(float* __restrict__ wr, const float* __restrict__ rowsum,
                         const float* __restrict__ wr_prev, const float* __restrict__ wlu,
                         const float* __restrict__ wu, const float* __restrict__ wg,
                         float* __restrict__ wu_new) {
    int t = blockIdx.x * blockDim.x + threadIdx.x;   // B*M
    int b = t >> 16;
    float sg = 1.0f / (1.0f + __expf(-wg[0]));
    float w  = wr[t] / rowsum[b];
    wr[t] = w;
    float ww = sg * wr_prev[t] + (1.0f - sg) * wlu[t];
    wu_new[t] = 0.5f * wu[t] + w + ww;
}

// ------------------------ r = wr @ memory_bank, split-K with f32 atomics
// 256 blocks (one per K-slab of 256). Block async-stages wr[0:32, kb:kb+256]
// into LDS once (reused by all 64 N-tiles), each wave computes both M-tiles
// for its 8 N-columns -> memory_bank read exactly once, coalesced B rows.
#define RG_KBLK 256
#define RG_STR  260     // floats; 1040B = 65*16: aligned rows, banks 4r+c
__global__ void k_rgemm(const float* __restrict__ wr, const float* __restrict__ mb,
                        float* __restrict__ r) {
    __shared__ float sA[32 * RG_STR];     // 33280 B
    int kb  = blockIdx.x * RG_KBLK;
    int tid = threadIdx.x;                // 0..255 (8 waves)

    // 32 rows x 256 floats = 2048 x 16B units; 8 per thread
    for (int i = 0; i < 8; ++i) {
        int u = i * 256 + tid;
        int row = u >> 6, c4 = (u & 63) * 4;
        async_ld_b128(wr + (size_t)row * M_ + kb + c4,
                      (unsigned int)(unsigned long long)&sA[row * RG_STR + c4]);
    }
    asm volatile("s_wait_asynccnt 0x0" ::: "memory");
    __syncthreads();

    int wv = tid >> 5, lane = tid & 31;
    int ln = lane & 15, hh = lane >> 4;
    for (int t = 0; t < 8; ++t) {
        int n0 = (wv * 8 + t) * 16;
        v8f acc0 = {}, acc1 = {};
        for (int k = 0; k < RG_KBLK; k += 4) {
            int ka = k + 2 * hh;
            v2f b; b.x = mb[(size_t)(kb + ka) * U_ + n0 + ln];
                   b.y = mb[(size_t)(kb + ka + 1) * U_ + n0 + ln];
            v2f a0 = *(const v2f*)(sA + ln * RG_STR + ka);
            v2f a1 = *(const v2f*)(sA + (16 + ln) * RG_STR + ka);
            acc0 = wmma_f32(a0, b, acc0);
            acc1 = wmma_f32(a1, b, acc1);
        }
        for (int j = 0; j < 8; ++j) {
            atomicAdd(&r[(j + 8 * hh) * U_ + n0 + ln], acc0[j]);
            atomicAdd(&r[(16 + j + 8 * hh) * U_ + n0 + ln], acc1[j]);
        }
    }
}

// ------------------------ two smallest per row + global argmin index
__global__ void k_select(const float* __restrict__ wu_new, float* __restrict__ nth,
                         int* __restrict__ sel) {
    __shared__ float sv1[256];
    __shared__ float sv2[256];
    __shared__ int   si1[256];
    int b = blockIdx.x;
    float v1 = 3.402823e38f, v2 = 3.402823e38f;
    int   i1 = 0x7fffffff;
    for (int m = threadIdx.x; m < M_; m += 256) {
        float v = wu_new[b * M_ + m];
        if (v < v1)       { v2 = v1; v1 = v; i1 = m; }
        else if (v == v1) { v2 = v1; }
        else if (v < v2)  { v2 = v; }
    }
    sv1[threadIdx.x] = v1; sv2[threadIdx.x] = v2; si1[threadIdx.x] = i1;
    __syncthreads();
    for (int o = 128; o > 0; o >>= 1) {
        if (threadIdx.x < o) {
            float av1 = sv1[threadIdx.x],     av2 = sv2[threadIdx.x];
            int   ai  = si1[threadIdx.x];
            float bv1 = sv1[threadIdx.x + o], bv2 = sv2[threadIdx.x + o];
            int   bi  = si1[threadIdx.x + o];
            if (bv1 < av1 || (bv1 == av1 && bi < ai)) {
                sv1[threadIdx.x] = bv1; si1[threadIdx.x] = bi;
                sv2[threadIdx.x] = fminf(bv2, av1);
            } else {
                sv2[threadIdx.x] = fminf(av2, bv1);
            }
        }
        __syncthreads();
    }
    if (threadIdx.x == 0) { nth[b] = sv2[0]; atomicMin(sel, si1[0]); }
}

// ------------------------ wlu_new = (wu_new <= nth[b])
__global__ void k_wlu(const float* __restrict__ wu_new, const float* __restrict__ nth,
                      float* __restrict__ wlu_new) {
    int t = blockIdx.x * blockDim.x + threadIdx.x;
    int b = t >> 16;
    wlu_new[t] = (wu_new[t] <= nth[b]) ? 1.0f : 0.0f;
}

// ------------------------ mem_new = ww.T @ h + mb * keep  (K = 32 outer product)
// 262144 waves: 4096 M-tiles x 64 N-tiles; all loads/stores coalesced
__global__ void k_memnew(const float* __restrict__ wr_prev, const float* __restrict__ wlu,
                         const float* __restrict__ h, const float* __restrict__ mb,
                         const float* __restrict__ wg, const int* __restrict__ sel,
                         float* __restrict__ mem_new) {
    int g    = (blockIdx.x * blockDim.x + threadIdx.x) >> 5;
    int lane = threadIdx.x & 31;
    int ln = lane & 15, hh = lane >> 4;
    int mt = g >> 6, nt = g & 63;
    int m0 = mt * 16, n0 = nt * 16;
    float sg = 1.0f / (1.0f + __expf(-wg[0]));
    int srow = *sel;
    v8f acc = {};
    for (int k = 0; k < B_; k += 4) {
        int ka = k + 2 * hh;
        v2f a;
        a.x = sg * wr_prev[(size_t)ka * M_ + m0 + ln] +
              (1.0f - sg) * wlu[(size_t)ka * M_ + m0 + ln];
        a.y = sg * wr_prev[(size_t)(ka + 1) * M_ + m0 + ln] +
              (1.0f - sg) * wlu[(size_t)(ka + 1) * M_ + m0 + ln];
        v2f b; b.x = h[ka * U_ + n0 + ln]; b.y = h[(ka + 1) * U_ + n0 + ln];
        acc = wmma_f32(a, b, acc);
    }
    for (int j = 0; j < 8; ++j) {
        int m = m0 + j + 8 * hh;
        float keep = (m == srow) ? 0.0f : 1.0f;
        mem_new[(size_t)m * U_ + n0 + ln] = acc[j] + mb[(size_t)m * U_ + n0 + ln] * keep;
    }
}

extern "C" void kernel_launch(void* const* d_in, const int* in_sizes, int n_in,
                              void* d_out, int out_size, void* d_ws, size_t ws_size,
                              hipStream_t stream) {
    (void)in_sizes; (void)n_in; (void)out_size; (void)ws_size;
    const float* x        = (const float*)d_in[0];
    const float* h_tm1    = (const float*)d_in[1];
    const float* c_tm1    = (const float*)d_in[2];
    const float* r_tm1    = (const float*)d_in[3];
    const float* mb       = (const float*)d_in[4];
    const float* wu       = (const float*)d_in[5];
    const float* wlu      = (const float*)d_in[6];
    const float* wr_prev  = (const float*)d_in[7];
    const float* W        = (const float*)d_in[8];
    const float* RK       = (const float*)d_in[9];
    const float* bias     = (const float*)d_in[10];
    const float* wg       = (const float*)d_in[11];

    float* out     = (float*)d_out;
    float* r_out   = out;                              // 32*1024
    float* h_out   = out + 32768;                      // 32*1024
    float* c_out   = out + 65536;                      // 32*1024
    float* mem_new = out + 98304;                      // 65536*1024
    float* wu_new  = out + 98304 + 67108864;           // 32*65536
    float* wlu_new = wu_new + 2097152;                 // 32*65536
    float* wr_out  = wlu_new + 2097152;                // 32*65536

    float* wsf    = (float*)d_ws;
    float* zcat   = wsf;                 // 32*4096
    float* ri     = wsf + 131072;        // 32*1024
    float* hinv   = wsf + 163840;        // 32
    float* rowsum = wsf + 163872;        // 32
    int*   sel    = (int*)(wsf + 163904);
    float* nth    = wsf + 163905;        // 32

    k_init  <<<128,   256, 0, stream>>>(r_out, rowsum, sel);
    k_gates <<<64,    256, 0, stream>>>(x, h_tm1, W, RK, bias, zcat);
    k_ri    <<<16,    256, 0, stream>>>(r_tm1, RK, ri);
    k_lstm  <<<128,   256, 0, stream>>>(zcat, ri, c_tm1, c_out, h_out);
    k_hnorm <<<32,    256, 0, stream>>>(h_out, hinv);
    k_sim   <<<4096,   64, 0, stream>>>(mb, h_out, hinv, wr_out, rowsum);
    k_normwu<<<8192,  256, 0, stream>>>(wr_out, rowsum, wr_prev, wlu, wu, wg, wu_new);
    k_rgemm <<<256,   256, 0, stream>>>(wr_out, mb, r_out);
    k_select<<<32,    256, 0, stream>>>(wu_new, nth, sel);
    k_wlu   <<<8192,  256, 0, stream>>>(wu_new, nth, wlu_new);
    k_memnew<<<32768, 256, 0, stream>>>(wr_prev, wlu, h_out, mb, wg, sel, mem_new);
}